// AttentionRecognitionHead_33895881900203
// MI455X (gfx1250) — compile-verified
//
#include <hip/hip_runtime.h>
#include <hip/hip_bf16.h>
#include <math.h>

// ---------------------------------------------------------------------------
// Problem constants (from reference)
// ---------------------------------------------------------------------------
#define BB   32
#define TT   512
#define XD   512
#define SD   512   // hidden
#define AD   512   // attention dim
#define NC   97
#define LLEN 25

typedef __attribute__((ext_vector_type(2))) float v2f;
typedef __attribute__((ext_vector_type(8))) float v8f;

// ---------------------------------------------------------------------------
// fp32 WMMA GEMM, C[m,n] = sum_k A[m,k]*B[k,n] + bias[n]
// One wave per 16x16 tile, K stepped by 4 via V_WMMA_F32_16X16X4_F32.
// A: row-major [M,lda]; B: row-major [K,ldb] ("nn")
// Inner loop is pointer-stepped: all clamping/base math hoisted out.
// ---------------------------------------------------------------------------
__global__ __launch_bounds__(32)
void gemm_nn_wmma(const float* __restrict__ A, int lda,
                  const float* __restrict__ Bm, int ldb,
                  const float* __restrict__ bias,
                  float* __restrict__ C, int ldc,
                  int M, int N, int K) {
  const int lane = threadIdx.x & 31;
  const int half = lane >> 4;      // 0: K pair {0,1}, 1: K pair {2,3}
  const int r    = lane & 15;
  const int n0   = blockIdx.x * 16;
  const int m0   = blockIdx.y * 16;

  int mA = m0 + r; if (mA > M - 1) mA = M - 1;   // clamp (no lane divergence)
  int nB = n0 + r; if (nB > N - 1) nB = N - 1;

  const float* __restrict__ ap = A  + (size_t)mA * lda + 2 * half; // contiguous pair
  const float* __restrict__ bp = Bm + (size_t)(2 * half) * ldb + nB;
  const size_t bstep = (size_t)4 * (size_t)ldb;

  v8f acc = {};
  #pragma unroll 4
  for (int k0 = 0; k0 < K; k0 += 4) {
    const v2f a = *reinterpret_cast<const v2f*>(ap);
    v2f b;
    b.x = bp[0];
    b.y = bp[ldb];
    acc = __builtin_amdgcn_wmma_f32_16x16x4_f32(
        /*neg_a=*/false, a, /*neg_b=*/false, b,
        /*c_mod=*/(short)0, acc, /*reuse_a=*/false, /*reuse_b=*/false);
    ap += 4;
    bp += bstep;
  }

  const float bn = bias ? bias[nB] : 0.0f;
  const int n = n0 + r;
  if (n < N) {
    float* __restrict__ cp = C + (size_t)(m0 + 8 * half) * ldc + n;
    #pragma unroll
    for (int i = 0; i < 8; ++i) {
      if (m0 + i + 8 * half < M) cp[(size_t)i * ldc] = acc[i] + bn;
    }
  }
}

// ---------------------------------------------------------------------------
// fp32 WMMA GEMM against transposed weights (torch layout):
// C[m,n] = sum_k A[m,k]*W[n,k] + bias[n];  W row-major [N, ldw=K]
// ---------------------------------------------------------------------------
__global__ __launch_bounds__(32)
void gemm_nt_wmma(const float* __restrict__ A, int lda,
                  const float* __restrict__ W, int ldw,
                  const float* __restrict__ bias,
                  float* __restrict__ C, int ldc,
                  int M, int N, int K) {
  const int lane = threadIdx.x & 31;
  const int half = lane >> 4;
  const int r    = lane & 15;
  const int n0   = blockIdx.x * 16;
  const int m0   = blockIdx.y * 16;

  int mA = m0 + r; if (mA > M - 1) mA = M - 1;
  int nB = n0 + r; if (nB > N - 1) nB = N - 1;   // clamp: OOB tiles still load in-bounds

  const float* __restrict__ ap = A + (size_t)mA * lda + 2 * half;
  const float* __restrict__ wp = W + (size_t)nB * ldw + 2 * half;

  v8f acc = {};
  #pragma unroll 4
  for (int k0 = 0; k0 < K; k0 += 4) {
    const v2f a = *reinterpret_cast<const v2f*>(ap);
    const v2f b = *reinterpret_cast<const v2f*>(wp);
    acc = __builtin_amdgcn_wmma_f32_16x16x4_f32(
        false, a, false, b, (short)0, acc, false, false);
    ap += 4;
    wp += 4;
  }

  const float bn = bias ? bias[nB] : 0.0f;
  const int n = n0 + r;
  if (n < N) {
    float* __restrict__ cp = C + (size_t)(m0 + 8 * half) * ldc + n;
    #pragma unroll
    for (int i = 0; i < 8; ++i) {
      if (m0 + i + 8 * half < M) cp[(size_t)i * ldc] = acc[i] + bn;
    }
  }
}

// ---------------------------------------------------------------------------
// Attention energies: e[b,t] = sum_a tanh(xProj[b,t,a] + sProj[b,a]) * w[a] + wb
// One wave32 per (b,t) row; 8 rows per 256-thread block.
// ---------------------------------------------------------------------------
__global__ __launch_bounds__(256)
void attn_e_kernel(const float* __restrict__ xProj,
                   const float* __restrict__ sProj,
                   const float* __restrict__ w,
                   const float* __restrict__ wb,
                   float* __restrict__ e) {
  const int wave = threadIdx.x >> 5;
  const int lane = threadIdx.x & 31;
  const int row  = blockIdx.x * 8 + wave;      // 0 .. B*T-1
  const int b    = row >> 9;                   // /T
  const float* xp = xProj + (size_t)row * AD;
  const float* sp = sProj + (size_t)b * AD;

  float acc = 0.0f;
  #pragma unroll 4
  for (int i = 0; i < AD / 32; ++i) {
    const int a = lane + i * 32;
    acc += tanhf(xp[a] + sp[a]) * w[a];
  }
  #pragma unroll
  for (int off = 16; off > 0; off >>= 1)
    acc += __shfl_xor(acc, off, 32);
  if (lane == 0) e[row] = acc + wb[0];
}

// ---------------------------------------------------------------------------
// Softmax over T=512 per batch row, in place. One block per b.
// ---------------------------------------------------------------------------
__global__ __launch_bounds__(256)
void softmax_kernel(float* __restrict__ alpha) {
  __shared__ float red[256];
  const int b = blockIdx.x, tid = threadIdx.x;
  float* a = alpha + (size_t)b * TT;
  const float v0 = a[tid], v1 = a[tid + 256];

  red[tid] = fmaxf(v0, v1);
  __syncthreads();
  for (int s = 128; s > 0; s >>= 1) {
    if (tid < s) red[tid] = fmaxf(red[tid], red[tid + s]);
    __syncthreads();
  }
  const float mx = red[0];
  __syncthreads();

  const float e0 = expf(v0 - mx), e1 = expf(v1 - mx);
  red[tid] = e0 + e1;
  __syncthreads();
  for (int s = 128; s > 0; s >>= 1) {
    if (tid < s) red[tid] += red[tid + s];
    __syncthreads();
  }
  const float inv = 1.0f / red[0];
  a[tid]       = e0 * inv;
  a[tid + 256] = e1 * inv;
}

// ---------------------------------------------------------------------------
// context[b,d] = sum_t alpha[b,t] * x[b,t,d]; also builds GRU input
// inp[b] = [ emb[y_prev[b]] (512) , context (512) ].  One block of 512 per b.
// ---------------------------------------------------------------------------
__global__ __launch_bounds__(512)
void context_kernel(const float* __restrict__ alpha,
                    const float* __restrict__ x,
                    const float* __restrict__ emb,
                    const int* __restrict__ targets,
                    int step,
                    float* __restrict__ inp) {
  const int b = blockIdx.x;
  const int d = threadIdx.x;
  const float* al = alpha + (size_t)b * TT;
  const float* xb = x + (size_t)b * TT * XD + d;

  float acc = 0.0f;
  #pragma unroll 4
  for (int t = 0; t < TT; ++t)
    acc += al[t] * xb[(size_t)t * XD];

  const int y = (step == 0) ? NC : targets[b * LLEN + (step - 1)];
  inp[(size_t)b * (AD + XD) + d]      = emb[(size_t)y * AD + d];
  inp[(size_t)b * (AD + XD) + AD + d] = acc;
}

// ---------------------------------------------------------------------------
// GRU gate update (in place on h). gi/gh are [B, 3H], gates ordered r,z,n.
// ---------------------------------------------------------------------------
__global__ __launch_bounds__(256)
void gru_update_kernel(const float* __restrict__ gi,
                       const float* __restrict__ gh,
                       float* __restrict__ h) {
  const int idx = blockIdx.x * blockDim.x + threadIdx.x;  // 0 .. B*SD-1
  const int b = idx >> 9;
  const int j = idx & (SD - 1);
  const float* gib = gi + (size_t)b * 3 * SD;
  const float* ghb = gh + (size_t)b * 3 * SD;

  const float r = 1.0f / (1.0f + expf(-(gib[j]          + ghb[j])));
  const float z = 1.0f / (1.0f + expf(-(gib[SD + j]     + ghb[SD + j])));
  const float n = tanhf(gib[2 * SD + j] + r * ghb[2 * SD + j]);
  h[idx] = (1.0f - z) * n + z * h[idx];
}

// ---------------------------------------------------------------------------
// Host launcher
// ---------------------------------------------------------------------------
extern "C" void kernel_launch(void* const* d_in, const int* in_sizes, int n_in,
                              void* d_out, int out_size, void* d_ws, size_t ws_size,
                              hipStream_t stream) {
  const float* x        = (const float*)d_in[0];   // [B,T,X]
  const int*   targets  = (const int*)  d_in[1];   // [B,25]
  const float* xEmbed_w = (const float*)d_in[2];   // [512,512]
  const float* xEmbed_b = (const float*)d_in[3];
  const float* sEmbed_w = (const float*)d_in[4];   // [512,512]
  const float* sEmbed_b = (const float*)d_in[5];
  const float* wEmbed_w = (const float*)d_in[6];   // [512,1]
  const float* wEmbed_b = (const float*)d_in[7];   // [1]
  const float* emb      = (const float*)d_in[8];   // [98,512]
  const float* gru_wih  = (const float*)d_in[9];   // [1536,1024]
  const float* gru_whh  = (const float*)d_in[10];  // [1536,512]
  const float* gru_bih  = (const float*)d_in[11];
  const float* gru_bhh  = (const float*)d_in[12];
  const float* fc_w     = (const float*)d_in[13];  // [97,512]
  const float* fc_b     = (const float*)d_in[14];
  float* out = (float*)d_out;                      // [B,25,97]

  // workspace layout (floats)
  float* ws    = (float*)d_ws;
  float* xProj = ws;                                  // B*T*A   = 8388608
  float* sProj = xProj + (size_t)BB * TT * AD;        // B*A     = 16384
  float* alpha = sProj + (size_t)BB * AD;             // B*T     = 16384
  float* inp   = alpha + (size_t)BB * TT;             // B*1024  = 32768
  float* gi    = inp   + (size_t)BB * (AD + XD);      // B*3H    = 49152
  float* gh    = gi    + (size_t)BB * 3 * SD;         // B*3H    = 49152
  float* h     = gh    + (size_t)BB * 3 * SD;         // B*H     = 16384

  // h0 = 0 (graph-capture-safe async memset)
  hipMemsetAsync(h, 0, (size_t)BB * SD * sizeof(float), stream);

  // One-time: xProj = x @ xEmbed_w + xEmbed_b   (M=16384, N=512, K=512)
  gemm_nn_wmma<<<dim3(AD / 16, (BB * TT) / 16), 32, 0, stream>>>(
      x, XD, xEmbed_w, AD, xEmbed_b, xProj, AD, BB * TT, AD, XD);

  for (int t = 0; t < LLEN; ++t) {
    // sProj = h @ sEmbed_w + sEmbed_b   (32 x 512 x 512)
    gemm_nn_wmma<<<dim3(AD / 16, BB / 16), 32, 0, stream>>>(
        h, SD, sEmbed_w, AD, sEmbed_b, sProj, AD, BB, AD, SD);

    // e = tanh(xProj + sProj) @ wEmbed_w + wb
    attn_e_kernel<<<(BB * TT) / 8, 256, 0, stream>>>(
        xProj, sProj, wEmbed_w, wEmbed_b, alpha);

    // alpha = softmax(e) over T
    softmax_kernel<<<BB, 256, 0, stream>>>(alpha);

    // context + GRU input assembly
    context_kernel<<<BB, 512, 0, stream>>>(alpha, x, emb, targets, t, inp);

    // gi = inp @ wih^T + bih   (32 x 1536 x 1024)
    gemm_nt_wmma<<<dim3((3 * SD) / 16, BB / 16), 32, 0, stream>>>(
        inp, AD + XD, gru_wih, AD + XD, gru_bih, gi, 3 * SD, BB, 3 * SD, AD + XD);

    // gh = h @ whh^T + bhh     (32 x 1536 x 512)
    gemm_nt_wmma<<<dim3((3 * SD) / 16, BB / 16), 32, 0, stream>>>(
        h, SD, gru_whh, SD, gru_bhh, gh, 3 * SD, BB, 3 * SD, SD);

    // h = GRU(h, gi, gh)
    gru_update_kernel<<<(BB * SD) / 256, 256, 0, stream>>>(gi, gh, h);

    // logits[:, t, :] = h @ fc_w^T + fc_b   (32 x 97 x 512, N padded to 112)
    gemm_nt_wmma<<<dim3((NC + 15) / 16, BB / 16), 32, 0, stream>>>(
        h, SD, fc_w, SD, fc_b, out + (size_t)t * NC, LLEN * NC, BB, NC, SD);
  }
}